// RepresentationSummarizer700Instance_30073361006608
// MI455X (gfx1250) — compile-verified
//
#include <hip/hip_runtime.h>
#include <hip/hip_bf16.h>

// Problem: R=700 independent Linear(1024 -> 1) over batch 128.
//   y[r,b] = sum_d S[r,b,d] * W[r,d] + bias[r];  out[b,r] = y[r,b]
// HBM-bound (AI ~ 0.5 FLOP/B). Keep f32, stream S once, compute with
// V_WMMA_F32_16X16X4_F32 (W replicated across the 16 N columns).

#define R_DIM 700
#define B_DIM 128
#define D_DIM 1024

typedef __attribute__((ext_vector_type(2))) float v2f;
typedef __attribute__((ext_vector_type(8))) float v8f;

__global__ void __launch_bounds__(256)
summarize700_wmma(const float* __restrict__ S,   // [700,128,1024]
                  const float* __restrict__ W,   // [700,1024]
                  const float* __restrict__ bias,// [700]
                  float* __restrict__ out)       // [128,700]
{
#if defined(__AMDGCN__)
    const int r    = blockIdx.x;          // residue, one workgroup each
    const int tid  = threadIdx.x;         // 256 threads = 8 wave32
    const int wave = tid >> 5;            // 0..7  -> batch tile of 16 rows
    const int lane = tid & 31;
    const int m     = lane & 15;          // A-matrix row within tile
    const int khalf = lane >> 4;          // 0: K=j, 1: K=j+2 (ISA A/B layout)

    // Stage W[r] (4 KB) into LDS once, coalesced float4 loads.
    __shared__ float w_lds[D_DIM];
    ((float4*)w_lds)[tid] = ((const float4*)(W + (size_t)r * D_DIM))[tid];
    __syncthreads();

    // Per-lane A stream pointer: row (wave*16 + m), starting at K offset 2*khalf.
    const float* __restrict__ aptr =
        S + (size_t)r * (B_DIM * D_DIM) + (size_t)(wave * 16 + m) * D_DIM + 2 * khalf;
    const float* __restrict__ wptr = w_lds + 2 * khalf;

    v8f acc = {};
    for (int k0 = 0; k0 < D_DIM; k0 += 32) {
        // Prefetch the A stream ~2KB ahead (speculative; OOB at tail is dropped).
        __builtin_prefetch(aptr + k0 + 512, 0, 1);
#pragma unroll
        for (int kk = 0; kk < 32; kk += 4) {
            // A tile 16x4: lane L, VGPR j <- S[row, k + j + 2*(L/16)]  (one b64 load)
            v2f a = *(const v2f*)(aptr + k0 + kk);
            // B tile 4x16: W replicated across all 16 N columns (LDS broadcast)
            v2f b = *(const v2f*)(wptr + k0 + kk);
            // (neg_a, A, neg_b, B, c_mod, C, reuse_a, reuse_b)
            acc = __builtin_amdgcn_wmma_f32_16x16x4_f32(
                false, a, false, b, (short)0, acc, false, false);
        }
    }

    // All 16 columns of D are identical. Column 0 lives in lane 0 (M=0..7)
    // and lane 16 (M=8..15), one row per accumulator VGPR.
    const float bb = bias[r];
    if (m == 0) {
        const int row0 = wave * 16 + 8 * khalf;
#pragma unroll
        for (int i = 0; i < 8; ++i) {
            out[(size_t)(row0 + i) * R_DIM + r] = acc[i] + bb;
        }
    }
#endif
}

extern "C" void kernel_launch(void* const* d_in, const int* in_sizes, int n_in,
                              void* d_out, int out_size, void* d_ws, size_t ws_size,
                              hipStream_t stream) {
    (void)in_sizes; (void)n_in; (void)out_size; (void)d_ws; (void)ws_size;
    const float* S    = (const float*)d_in[0];   // s_s [700,128,1024] f32
    const float* W    = (const float*)d_in[1];   // W   [700,1024,1]  f32
    const float* bias = (const float*)d_in[2];   // b   [700,1]       f32
    float* out        = (float*)d_out;           // [128,700,1]       f32

    summarize700_wmma<<<dim3(R_DIM), dim3(256), 0, stream>>>(S, W, bias, out);
}